// VectorQuantizer_CosSim_1271310319902
// MI455X (gfx1250) — compile-verified
//
#include <hip/hip_runtime.h>
#include <math.h>

// Problem sizes (fixed by the reference)
#define MROWS 25088      // B*N = 128*196
#define KCIN  768
#define KDIM  64
#define NEMB  8192
#define NSEG  8
#define SEGSZ (NEMB / NSEG)   // 1024 embeddings per argmax segment

typedef __attribute__((ext_vector_type(2))) float v2f;
typedef __attribute__((ext_vector_type(8))) float v8f;

// D(16x16,f32) = A(16x4,f32) * B(4x16,f32) + C  — CDNA5 fp32 WMMA
__device__ __forceinline__ v8f wmma4(v2f a, v2f b, v8f c) {
    return __builtin_amdgcn_wmma_f32_16x16x4_f32(
        /*neg_a=*/false, a, /*neg_b=*/false, b,
        /*c_mod=*/(short)0, c, /*reuse_a=*/false, /*reuse_b=*/false);
}

// ---------------------------------------------------------------- k_prep ----
// Per-row L2 norm of the codebook (8192 x 64).
__global__ void k_prep(const float* __restrict__ emb, float* __restrict__ enorm) {
    int e = blockIdx.x * blockDim.x + threadIdx.x;
    if (e < NEMB) {
        const float4* p = (const float4*)(emb + (size_t)e * KDIM);
        float s = 0.f;
#pragma unroll
        for (int t = 0; t < KDIM / 4; ++t) {
            float4 q = p[t];
            s += q.x * q.x + q.y * q.y + q.z * q.z + q.w * q.w;
        }
        enorm[e] = sqrtf(s);
    }
}

// ------------------------------------------------------------ k_compress ----
// zf[25088,64] = z[25088,768] @ Wc^T[768,64] + bc ; also per-row ||zf||^2.
// One wave -> 16 M-rows x all 64 D-cols (4 accumulators). K stepped by 4.
__global__ void k_compress(const float* __restrict__ z, const float* __restrict__ Wc,
                           const float* __restrict__ bc, float* __restrict__ zf,
                           float* __restrict__ zfn2) {
    const int gw   = blockIdx.x * (blockDim.x >> 5) + (threadIdx.x >> 5); // 0..1567
    const int lane = threadIdx.x & 31;
    const int hf   = lane >> 4;      // K-half selector
    const int lr   = lane & 15;
    const int m0   = gw * 16;

    v8f acc0 = 0.f, acc1 = 0.f, acc2 = 0.f, acc3 = 0.f;
    const float* zrow = z + (size_t)(m0 + lr) * KCIN + 2 * hf;
    const float* w0   = Wc + (size_t)( 0 + lr) * KCIN + 2 * hf;
    const float* w1   = Wc + (size_t)(16 + lr) * KCIN + 2 * hf;
    const float* w2   = Wc + (size_t)(32 + lr) * KCIN + 2 * hf;
    const float* w3   = Wc + (size_t)(48 + lr) * KCIN + 2 * hf;

#pragma unroll 4
    for (int k = 0; k < KCIN; k += 4) {
        v2f a = *(const v2f*)(zrow + k);
        acc0 = wmma4(a, *(const v2f*)(w0 + k), acc0);
        acc1 = wmma4(a, *(const v2f*)(w1 + k), acc1);
        acc2 = wmma4(a, *(const v2f*)(w2 + k), acc2);
        acc3 = wmma4(a, *(const v2f*)(w3 + k), acc3);
    }

    // bias + store + per-row norm^2 (row m0+v+8*hf, column t*16+lr)
    float nrm[8];
#pragma unroll
    for (int v = 0; v < 8; ++v) nrm[v] = 0.f;

    v8f* accs[4] = { &acc0, &acc1, &acc2, &acc3 };
#pragma unroll
    for (int t = 0; t < 4; ++t) {
        float bias = bc[t * 16 + lr];
#pragma unroll
        for (int v = 0; v < 8; ++v) {
            float val = (*accs[t])[v] + bias;
            zf[(size_t)(m0 + v + 8 * hf) * KDIM + t * 16 + lr] = val;
            nrm[v] += val * val;
        }
    }
#pragma unroll
    for (int v = 0; v < 8; ++v) {
        float s = nrm[v];
        s += __shfl_xor(s, 1, 32);
        s += __shfl_xor(s, 2, 32);
        s += __shfl_xor(s, 4, 32);
        s += __shfl_xor(s, 8, 32);   // xor<16 stays inside each 16-lane half
        if (lr == 0) zfn2[m0 + v + 8 * hf] = s;
    }
}

// ------------------------------------------------------------- k_argmax ----
// Fused sim = zf @ emb^T with running argmax over one NE segment.
// Wave handles 16 rows x 1024 embeddings (64 WMMA tiles, K=64 in 16 steps).
__global__ void k_argmax(const float* __restrict__ zf, const float* __restrict__ emb,
                         float* __restrict__ bestv, int* __restrict__ besti) {
    const int gw   = blockIdx.x * (blockDim.x >> 5) + (threadIdx.x >> 5); // m-tile
    const int seg  = blockIdx.y;
    const int lane = threadIdx.x & 31;
    const int hf   = lane >> 4;
    const int lr   = lane & 15;
    const int m0   = gw * 16;

    // Preload this wave's A tile for the whole segment loop (K=64 -> 16 x v2f)
    v2f a[16];
    const float* zr = zf + (size_t)(m0 + lr) * KDIM + 2 * hf;
#pragma unroll
    for (int t = 0; t < 16; ++t) a[t] = *(const v2f*)(zr + 4 * t);

    float bv[8];
    int   bi[8];
#pragma unroll
    for (int v = 0; v < 8; ++v) { bv[v] = -3.402823466e38f; bi[v] = 0; }

    const int e0base = seg * SEGSZ;
    for (int et = 0; et < SEGSZ / 16; ++et) {
        const int e0 = e0base + et * 16;
        v8f d = 0.f;
        const float* er = emb + (size_t)(e0 + lr) * KDIM + 2 * hf;
#pragma unroll
        for (int t = 0; t < 16; ++t)
            d = wmma4(a[t], *(const v2f*)(er + 4 * t), d);
        const int e = e0 + lr;   // this lane's column
#pragma unroll
        for (int v = 0; v < 8; ++v) {
            float val = d[v];
            if (val > bv[v]) { bv[v] = val; bi[v] = e; }  // strict > keeps first
        }
    }

    // Reduce across the 16 lanes of each half (row m0+v+8*hf)
#pragma unroll
    for (int v = 0; v < 8; ++v) {
        float val = bv[v];
        int   idx = bi[v];
#pragma unroll
        for (int msk = 1; msk < 16; msk <<= 1) {
            float ov = __shfl_xor(val, msk, 32);
            int   oi = __shfl_xor(idx, msk, 32);
            if (ov > val || (ov == val && oi < idx)) { val = ov; idx = oi; }
        }
        if (lr == 0) {
            int row = m0 + v + 8 * hf;
            bestv[(size_t)row * NSEG + seg] = val;
            besti[(size_t)row * NSEG + seg] = idx;
        }
    }
}

// ------------------------------------------------------------- k_select ----
// Combine segment candidates, compute (1 - cos) per row, deterministic
// per-block partial sums (no float atomics).
__global__ void k_select(const float* __restrict__ bestv, const int* __restrict__ besti,
                         const float* __restrict__ zfn2, const float* __restrict__ enorm,
                         int* __restrict__ idx_out, float* __restrict__ partials) {
    __shared__ float red[256];
    int m = blockIdx.x * blockDim.x + threadIdx.x;
    float contrib = 0.f;
    if (m < MROWS) {
        float bv = -3.402823466e38f;
        int   bi = 0;
#pragma unroll
        for (int s = 0; s < NSEG; ++s) {        // ascending e => strict > keeps first
            float v = bestv[(size_t)m * NSEG + s];
            int   i = besti[(size_t)m * NSEG + s];
            if (v > bv) { bv = v; bi = i; }
        }
        idx_out[m] = bi;
        float zn = fmaxf(sqrtf(zfn2[m]), 1e-8f);
        float en = fmaxf(enorm[bi], 1e-8f);
        contrib = 1.0f - bv / (zn * en);
    }
    red[threadIdx.x] = contrib;
    __syncthreads();
    for (int s = 128; s > 0; s >>= 1) {
        if (threadIdx.x < s) red[threadIdx.x] += red[threadIdx.x + s];
        __syncthreads();
    }
    if (threadIdx.x == 0) partials[blockIdx.x] = red[0];
}

// ------------------------------------------------------------- k_expand ----
// out[25088,768] = emb[idx] @ We^T + be.  Gather fused into A-load.
// Wave -> 16 rows x 64 cols (blockIdx.y picks which 64-col group of 768).
__global__ void k_expand(const float* __restrict__ emb, const int* __restrict__ idx,
                         const float* __restrict__ We, const float* __restrict__ be,
                         float* __restrict__ out) {
    const int gw   = blockIdx.x * (blockDim.x >> 5) + (threadIdx.x >> 5); // m-tile
    const int ng   = blockIdx.y;                                          // 0..11
    const int lane = threadIdx.x & 31;
    const int hf   = lane >> 4;
    const int lr   = lane & 15;
    const int m0   = gw * 16;

    const int r = idx[m0 + lr];                 // gathered codebook row
    v2f a[16];
    const float* ar = emb + (size_t)r * KDIM + 2 * hf;
#pragma unroll
    for (int t = 0; t < 16; ++t) a[t] = *(const v2f*)(ar + 4 * t);

#pragma unroll
    for (int nt = 0; nt < 4; ++nt) {
        const int n0 = ng * 64 + nt * 16;
        v8f d = 0.f;
        const float* wr = We + (size_t)(n0 + lr) * KDIM + 2 * hf;
#pragma unroll
        for (int t = 0; t < 16; ++t)
            d = wmma4(a[t], *(const v2f*)(wr + 4 * t), d);
        float bias = be[n0 + lr];
#pragma unroll
        for (int v = 0; v < 8; ++v)
            out[(size_t)(m0 + v + 8 * hf) * KCIN + n0 + lr] = d[v] + bias;
    }
}

// -------------------------------------------------------------- k_final ----
__global__ void k_final(const float* __restrict__ partials, int nparts,
                        float* __restrict__ loss_out) {
    if (threadIdx.x == 0 && blockIdx.x == 0) {
        float s = 0.f;
        for (int i = 0; i < nparts; ++i) s += partials[i];
        // forward value of both cosine terms is identical -> 3 * mean(1-cos)
        *loss_out = 3.0f * s / (float)MROWS;
    }
}

// ---------------------------------------------------------------------------
extern "C" void kernel_launch(void* const* d_in, const int* in_sizes, int n_in,
                              void* d_out, int out_size, void* d_ws, size_t ws_size,
                              hipStream_t stream) {
    const float* z   = (const float*)d_in[0];  // [128,196,768]
    const float* emb = (const float*)d_in[1];  // [8192,64]
    const float* Wc  = (const float*)d_in[2];  // [64,768]
    const float* bc  = (const float*)d_in[3];  // [64]
    const float* We  = (const float*)d_in[4];  // [768,64]
    const float* be  = (const float*)d_in[5];  // [768]
    float* out = (float*)d_out;                // [25088*768] ++ [1] loss

    char* ws = (char*)d_ws;
    float* zf    = (float*)(ws);               // 25088*64   (6,422,528 B)
    float* zfn2  = (float*)(ws + 6422528);     // 25088      (  100,352 B)
    float* enorm = (float*)(ws + 6522880);     // 8192       (   32,768 B)
    float* bestv = (float*)(ws + 6555648);     // 25088*8    (  802,816 B)
    int*   besti = (int*)  (ws + 7358464);     // 25088*8    (  802,816 B)
    int*   idx   = (int*)  (ws + 8161280);     // 25088      (  100,352 B)
    float* part  = (float*)(ws + 8261632);     // 98 partial sums

    k_prep    <<<dim3(NEMB / 256),  dim3(256), 0, stream>>>(emb, enorm);
    k_compress<<<dim3(196),         dim3(256), 0, stream>>>(z, Wc, bc, zf, zfn2);
    k_argmax  <<<dim3(196, NSEG),   dim3(256), 0, stream>>>(zf, emb, bestv, besti);
    k_select  <<<dim3(98),          dim3(256), 0, stream>>>(bestv, besti, zfn2, enorm, idx, part);
    k_expand  <<<dim3(196, 12),     dim3(256), 0, stream>>>(emb, idx, We, be, out);
    k_final   <<<dim3(1),           dim3(1),   0, stream>>>(part, 98, out + (size_t)MROWS * KCIN);
}